// CausalWindowedAttention_36747740184645
// MI455X (gfx1250) — compile-verified
//
#include <hip/hip_runtime.h>

// CDNA5 / gfx1250 windowed causal attention.
// out  = softmax(mask(QK^T / 8)) @ V          (B,H,S,D) = (2,16,2048,64)
// attn = softmax(mask(QK^T / 8))              (B,H,S,S), window = 256, causal.
//
// Roofline: ~512MB attn stores dominate -> ~25us floor at 23.3 TB/s HBM.
// All bulk traffic (attn stores, Q/K/V staging, softmax LDS passes) is b128.
// fp32 WMMA (V_WMMA_F32_16X16X4_F32) keeps reference numerics; compute is
// negligible vs. the attn materialization.

typedef __attribute__((ext_vector_type(2))) float v2f;
typedef __attribute__((ext_vector_type(8))) float v8f;

#define S_LEN    2048
#define D_DIM    64
#define WINDOW   256
#define NEG_BIG  (-1.0e9f)
#define TEMP_INV (0.125f)
#define QK_STRIDE 68    // 272B rows: 16B-aligned, bank stride 4 (conflict-free)
#define SS_COLS   276   // >= 17*16=272 band cols; 16B-aligned rows, 20*m mod 64 distinct

union V8U { v8f v; float f[8]; };

__global__ __launch_bounds__(32)
void attn_win_kernel(const float* __restrict__ q,
                     const float* __restrict__ k,
                     const float* __restrict__ v,
                     float* __restrict__ out,
                     float* __restrict__ attn) {
  __shared__ float sQ[16][QK_STRIDE];
  __shared__ float sKV[16][QK_STRIDE];   // K tiles in phase 1, V tiles in phase 4
  __shared__ float sS[16][SS_COLS];

  const int lane = threadIdx.x;        // 0..31 (wave32)
  const int qt   = blockIdx.x;         // query tile index (16 rows)
  const int bh   = blockIdx.y;         // b*H + h
  const int i0   = qt * 16;

  const float* Qb = q + ((size_t)bh * S_LEN + i0) * D_DIM;
  const float* Kb = k + (size_t)bh * S_LEN * D_DIM;
  const float* Vb = v + (size_t)bh * S_LEN * D_DIM;

  // ---- stage Q tile (16x64) into LDS via b128 ----
  for (int it = 0; it < 8; ++it) {
    int linear = it * 32 + lane;            // 0..255 float4 groups
    int r  = linear >> 4;
    int c4 = (linear & 15) * 4;
    *(float4*)&sQ[r][c4] = *(const float4*)&Qb[(size_t)r * D_DIM + c4];
  }

  // key window: j in (i-256, i], tiles aligned to 16
  int jlo = i0 - (WINDOW - 1); if (jlo < 0) jlo = 0;
  const int t0     = jlo & ~15;
  const int ntiles = ((i0 + 15) - t0) / 16 + 1;   // <= 17
  const int ncols  = ntiles * 16;                 // multiple of 16

  // WMMA f32 16x16x4 operand lane mapping:
  //  A (16x4): lanes 0-15 -> M=lane, K=0,1 in {a.x,a.y}; lanes 16-31 -> K=2,3
  //  B (4x16): lanes 0-15 -> N=lane, K=0,1 in {b.x,b.y}; lanes 16-31 -> K=2,3
  //  C (16x16): VGPR r -> M=r (lanes 0-15) / M=r+8 (lanes 16-31); N = lane&15
  const int mn    = lane & 15;
  const int koff  = (lane >> 4) * 2;
  const int mbase = (lane >> 4) * 8;

  __syncthreads();

  // ---- phase 1: scores S = mask(Q K^T / 8) into LDS strip ----
  for (int kt = 0; kt < ntiles; ++kt) {
    const int key0 = t0 + kt * 16;
    const float* Kt = Kb + (size_t)key0 * D_DIM;

    for (int it = 0; it < 8; ++it) {
      int linear = it * 32 + lane;
      int r  = linear >> 4;
      int c4 = (linear & 15) * 4;
      *(float4*)&sKV[r][c4] = *(const float4*)&Kt[(size_t)r * D_DIM + c4];
    }
    // prefetch next K tile (global_prefetch_b8)
    if (kt + 1 < ntiles) {
      const float* Kn = Kb + (size_t)(key0 + 16) * D_DIM;
      __builtin_prefetch((const void*)(Kn + lane * 32), 0, 3);
    }
    __syncthreads();

    v8f c8 = {};
    for (int kb = 0; kb < D_DIM; kb += 4) {
      v2f a, b;
      a.x = sQ[mn][kb + koff];
      a.y = sQ[mn][kb + koff + 1];
      b.x = sKV[mn][kb + koff];     // B[k][n] = K[n][k]
      b.y = sKV[mn][kb + koff + 1];
      c8 = __builtin_amdgcn_wmma_f32_16x16x4_f32(
              false, a, false, b, (short)0, c8, false, false);
    }

    V8U cu; cu.v = c8;
    const int kj = key0 + mn;               // global key index
    for (int r = 0; r < 8; ++r) {
      int m  = r + mbase;
      int qi = i0 + m;                      // global query index
      bool ok = (kj <= qi) && (kj > qi - WINDOW);
      sS[m][kt * 16 + mn] = ok ? (cu.f[r] * TEMP_INV) : NEG_BIG;
    }
    __syncthreads();
  }

  // ---- phase 2: row softmax in LDS (lanes 0-15, one row each; b128) ----
  if (lane < 16) {
    const int ng = ncols >> 2;
    float mx = -3.4e38f;
    for (int g = 0; g < ng; ++g) {
      float4 t = *(float4*)&sS[lane][g * 4];
      mx = fmaxf(mx, fmaxf(fmaxf(t.x, t.y), fmaxf(t.z, t.w)));
    }
    float sum = 0.0f;
    for (int g = 0; g < ng; ++g) {
      float4 t = *(float4*)&sS[lane][g * 4];
      t.x = __expf(t.x - mx);   // masked: exp(-1e9-..) -> 0
      t.y = __expf(t.y - mx);
      t.z = __expf(t.z - mx);
      t.w = __expf(t.w - mx);
      sum += (t.x + t.y) + (t.z + t.w);
      *(float4*)&sS[lane][g * 4] = t;
    }
    float inv = 1.0f / sum;
    for (int g = 0; g < ng; ++g) {
      float4 t = *(float4*)&sS[lane][g * 4];
      t.x *= inv; t.y *= inv; t.z *= inv; t.w *= inv;
      *(float4*)&sS[lane][g * 4] = t;
    }
  }
  __syncthreads();

  // ---- phase 3: write FULL attn rows, b128 (band from LDS, zeros outside) --
  // t0 and ncols are multiples of 16 => each float4 group is entirely
  // inside or outside the band. One pass covers the row: no memset needed.
  {
    float* Ab = attn + ((size_t)bh * S_LEN + i0) * S_LEN;
    for (int m = 0; m < 16; ++m) {
      float4* row4 = (float4*)(Ab + (size_t)m * S_LEN);
      for (int g = lane; g < (S_LEN / 4); g += 32) {
        int rel = g * 4 - t0;
        float4 val;
        if (rel >= 0 && rel < ncols) val = *(float4*)&sS[m][rel];
        else                         val = make_float4(0.f, 0.f, 0.f, 0.f);
        row4[g] = val;
      }
    }
  }

  // ---- phase 4: out = P @ V (f32 WMMA, 4 N-tiles over D=64) ----
  v8f acc0 = {}, acc1 = {}, acc2 = {}, acc3 = {};
  for (int kt = 0; kt < ntiles; ++kt) {
    const int key0 = t0 + kt * 16;
    const float* Vt = Vb + (size_t)key0 * D_DIM;

    __syncthreads();   // previous sKV consumers done before overwrite
    for (int it = 0; it < 8; ++it) {
      int linear = it * 32 + lane;
      int r  = linear >> 4;
      int c4 = (linear & 15) * 4;
      *(float4*)&sKV[r][c4] = *(const float4*)&Vt[(size_t)r * D_DIM + c4];
    }
    if (kt + 1 < ntiles) {
      const float* Vn = Vb + (size_t)(key0 + 16) * D_DIM;
      __builtin_prefetch((const void*)(Vn + lane * 32), 0, 3);
    }
    __syncthreads();

    for (int kk = 0; kk < 4; ++kk) {
      const int kbase = kt * 16 + kk * 4;
      v2f a;
      a.x = sS[mn][kbase + koff];
      a.y = sS[mn][kbase + koff + 1];
      const int vr = kk * 4 + koff;
      v2f b0, b1, b2, b3;
      b0.x = sKV[vr][ 0 + mn]; b0.y = sKV[vr + 1][ 0 + mn];
      b1.x = sKV[vr][16 + mn]; b1.y = sKV[vr + 1][16 + mn];
      b2.x = sKV[vr][32 + mn]; b2.y = sKV[vr + 1][32 + mn];
      b3.x = sKV[vr][48 + mn]; b3.y = sKV[vr + 1][48 + mn];
      acc0 = __builtin_amdgcn_wmma_f32_16x16x4_f32(false, a, false, b0, (short)0, acc0, false, false);
      acc1 = __builtin_amdgcn_wmma_f32_16x16x4_f32(false, a, false, b1, (short)0, acc1, false, false);
      acc2 = __builtin_amdgcn_wmma_f32_16x16x4_f32(false, a, false, b2, (short)0, acc2, false, false);
      acc3 = __builtin_amdgcn_wmma_f32_16x16x4_f32(false, a, false, b3, (short)0, acc3, false, false);
    }
  }

  // ---- store out tile (16x64) ----
  {
    float* Ob = out + ((size_t)bh * S_LEN + i0) * D_DIM;
    V8U u0, u1, u2, u3;
    u0.v = acc0; u1.v = acc1; u2.v = acc2; u3.v = acc3;
    for (int r = 0; r < 8; ++r) {
      int m = r + mbase;
      float* row = Ob + (size_t)m * D_DIM;
      row[ 0 + mn] = u0.f[r];
      row[16 + mn] = u1.f[r];
      row[32 + mn] = u2.f[r];
      row[48 + mn] = u3.f[r];
    }
  }
}

extern "C" void kernel_launch(void* const* d_in, const int* in_sizes, int n_in,
                              void* d_out, int out_size, void* d_ws, size_t ws_size,
                              hipStream_t stream) {
  (void)in_sizes; (void)n_in; (void)out_size; (void)d_ws; (void)ws_size;
  const float* q = (const float*)d_in[0];
  const float* k = (const float*)d_in[1];
  const float* v = (const float*)d_in[2];

  float* out = (float*)d_out;
  const size_t out_elems = (size_t)2 * 16 * S_LEN * D_DIM;  // 4,194,304
  float* attn = out + out_elems;                            // 134,217,728 elems

  dim3 grid(S_LEN / 16, 2 * 16);   // (128 q-tiles, B*H=32)
  attn_win_kernel<<<grid, 32, 0, stream>>>(q, k, v, out, attn);
}